// HGATConv_64510408786253
// MI455X (gfx1250) — compile-verified
//
#include <hip/hip_runtime.h>
#include <math.h>

typedef __attribute__((ext_vector_type(2))) float v2f;
typedef __attribute__((ext_vector_type(8))) float v8f;

static constexpr float MIN_NORM_C  = 1e-15f;
static constexpr float MAXNORM_C   = 1.0f - 4e-3f;   // (1-EPS)/sqrt(c), c=1
static constexpr float ATANH_EPS   = 1e-7f;
static constexpr float NEG_ATT     = 0.2f;
static constexpr float NEG_ACT     = 0.01f;

__device__ __forceinline__ float artanh_c(float v) {
  v = fminf(v, 1.0f - ATANH_EPS);
  v = fmaxf(v, -1.0f + ATANH_EPS);
  return 0.5f * logf((1.0f + v) / (1.0f - v));
}

__device__ __forceinline__ float wred(float v) {   // full wave32 sum
#pragma unroll
  for (int m = 16; m; m >>= 1) v += __shfl_xor(v, m, 32);
  return v;
}
__device__ __forceinline__ float hred(float v) {   // 16-lane half-wave sum
#pragma unroll
  for (int m = 8; m; m >>= 1) v += __shfl_xor(v, m, 32);
  return v;
}

// ---------------- hyp_bias = proj(expmap0(bias)) + its squared norm ----------
__global__ void k_bias(const float* __restrict__ bias, float* __restrict__ hb) {
  __shared__ float red[128];
  int t = threadIdx.x;                   // 128 threads
  float b = bias[t];
  red[t] = b * b;
  __syncthreads();
  for (int s = 64; s > 0; s >>= 1) { if (t < s) red[t] += red[t + s]; __syncthreads(); }
  float n  = fmaxf(sqrtf(red[0]), MIN_NORM_C);
  __syncthreads();
  float v  = tanhf(n) / n * b;                       // expmap0
  float nn = fmaxf(tanhf(n), MIN_NORM_C);            // norm after expmap0
  float p  = nn > MAXNORM_C ? MAXNORM_C / nn : 1.0f; // proj
  v *= p;
  red[t] = v * v;
  __syncthreads();
  for (int s = 64; s > 0; s >>= 1) { if (t < s) red[t] += red[t + s]; __syncthreads(); }
  hb[t] = v;
  if (t == 0) hb[128] = red[0];
}

// ---------------- mx = x @ W^T via V_WMMA_F32_16X16X4_F32 --------------------
// Block = 256 thr = 8 waves; block handles 16 rows of x, wave w -> N-tile w.
// 32-bit A 16x4 layout: lanes 0-15 K={0,1}, lanes 16-31 K={2,3} (VGPR0=K, VGPR1=K+1)
// 32-bit B 4x16 layout mirrors it with N across lanes -> both are float2 loads.
__global__ __launch_bounds__(256) void k_matmul(const float* __restrict__ x,
                                                const float* __restrict__ w,
                                                float* __restrict__ mx, int N) {
  int wave = threadIdx.x >> 5;
  int lane = threadIdx.x & 31;
  int half = lane >> 4, l16 = lane & 15;
  int rowBase = blockIdx.x * 16;
  int nBase   = wave * 16;
  int row = rowBase + l16; if (row > N - 1) row = N - 1;
  const float* xrow = x + (size_t)row * 128;
  const float* wrow = w + (size_t)(nBase + l16) * 128;   // B[k][n] = W[n][k]
  v8f acc = {};
#pragma unroll 4
  for (int k = 0; k < 128; k += 4) {
    int kk = k + 2 * half;
    v2f a = *(const v2f*)(xrow + kk);
    v2f b = *(const v2f*)(wrow + kk);
    acc = __builtin_amdgcn_wmma_f32_16x16x4_f32(false, a, false, b,
                                                (short)0, acc, false, false);
  }
  // C/D layout: lane l, VGPR v -> (M = v + 8*half, N = l16)
  float* mrow = mx + (size_t)rowBase * 128 + nBase + l16;
  if (rowBase + 16 <= N) {               // fast path: whole tile in range
#pragma unroll
    for (int v = 0; v < 8; ++v) mrow[(size_t)(v + 8 * half) * 128] = acc[v];
  } else {                               // ragged last block only
#pragma unroll
    for (int v = 0; v < 8; ++v) {
      int m = v + 8 * half;
      if (rowBase + m < N) mrow[(size_t)m * 128] = acc[v];
    }
  }
}

// ------- per node: mobius_matvec -> proj -> mobius_add(bias) -> proj ---------
// ------- -> logmap0 -> xt, per-head attention logits a_i, a_j ----------------
__global__ __launch_bounds__(256) void k_node(const float* __restrict__ x,
    const float* __restrict__ mx, const float* __restrict__ hb,
    const float* __restrict__ atti, const float* __restrict__ attj,
    float* __restrict__ xt, float* __restrict__ ai, float* __restrict__ aj, int N) {
  int node = blockIdx.x * 8 + (threadIdx.x >> 5);
  if (node >= N) return;
  int lane = threadIdx.x & 31;
  float4 xv = ((const float4*)(x  + (size_t)node * 128))[lane];
  float4 mv = ((const float4*)(mx + (size_t)node * 128))[lane];
  float4 hv = ((const float4*)hb)[lane];
  float hb2 = hb[128];

  float x2 = wred(xv.x*xv.x + xv.y*xv.y + xv.z*xv.z + xv.w*xv.w);
  float m2 = wred(mv.x*mv.x + mv.y*mv.y + mv.z*mv.z + mv.w*mv.w);
  float xn = fmaxf(sqrtf(x2), MIN_NORM_C);
  float mn = fmaxf(sqrtf(m2), MIN_NORM_C);
  // mobius_matvec (c=1): tanh(||mx||/||x|| * artanh(||x||)) * mx / ||mx||
  float s  = tanhf(mn / xn * artanh_c(xn)) / mn;
  float rx = mv.x*s, ry = mv.y*s, rz = mv.z*s, rw = mv.w*s;
  // proj
  float rn = fmaxf(fabsf(s) * mn, MIN_NORM_C);
  float p1 = rn > MAXNORM_C ? MAXNORM_C / rn : 1.0f;
  rx *= p1; ry *= p1; rz *= p1; rw *= p1;
  // mobius_add(h, hyp_bias)
  float h2 = wred(rx*rx + ry*ry + rz*rz + rw*rw);
  float xy = wred(rx*hv.x + ry*hv.y + rz*hv.z + rw*hv.w);
  float ca = 1.0f + 2.0f*xy + hb2;
  float cb = 1.0f - h2;
  float iv = 1.0f / fmaxf(1.0f + 2.0f*xy + h2*hb2, MIN_NORM_C);
  float gx = (ca*rx + cb*hv.x)*iv, gy = (ca*ry + cb*hv.y)*iv;
  float gz = (ca*rz + cb*hv.z)*iv, gw = (ca*rw + cb*hv.w)*iv;
  // proj
  float g2 = wred(gx*gx + gy*gy + gz*gz + gw*gw);
  float gn = fmaxf(sqrtf(g2), MIN_NORM_C);
  float p2 = gn > MAXNORM_C ? MAXNORM_C / gn : 1.0f;
  gx *= p2; gy *= p2; gz *= p2; gw *= p2;
  // logmap0
  float nn  = fmaxf(gn * p2, MIN_NORM_C);
  float lsc = artanh_c(nn) / nn;
  float tx = gx*lsc, ty = gy*lsc, tz = gz*lsc, tw = gw*lsc;
  ((float4*)(xt + (size_t)node * 128))[lane] = make_float4(tx, ty, tz, tw);
  // per-head attention logits: lane covers channels 4*lane..4*lane+3
  int head = lane >> 4;
  int cin  = (lane & 15) * 4;
  const float* aih = atti + head * 64 + cin;
  const float* ajh = attj + head * 64 + cin;
  float di = tx*aih[0] + ty*aih[1] + tz*aih[2] + tw*aih[3];
  float dj = tx*ajh[0] + ty*ajh[1] + tz*ajh[2] + tw*ajh[3];
  di = hred(di); dj = hred(dj);
  if ((lane & 15) == 0) { ai[node*2 + head] = di; aj[node*2 + head] = dj; }
}

// ---------------- zero supp/den, amax = -inf ---------------------------------
__global__ void k_init(float* __restrict__ supp, float* __restrict__ den,
                       float* __restrict__ amax, int N) {
  int i = blockIdx.x * blockDim.x + threadIdx.x;
  if (i < N * 128) supp[i] = 0.0f;
  if (i < N * 2) { den[i] = 0.0f; amax[i] = -__builtin_inff(); }
}

__device__ __forceinline__ void edge_ids(int t, int E, const long long* src,
                                         const long long* dst, int& ei, int& ej) {
  if (t < E) { ei = (int)src[t]; ej = (int)dst[t]; }
  else       { ei = t - E;       ej = t - E; }
}

// ---------------- segment max of leaky(a_i[ei]+a_j[ej]) ----------------------
__global__ void k_edge_max(const long long* __restrict__ src, const long long* __restrict__ dst,
                           const float* __restrict__ ai, const float* __restrict__ aj,
                           float* __restrict__ amax, int E, int N) {
  int t = blockIdx.x * blockDim.x + threadIdx.x;
  if (t >= E + N) return;
  int ei, ej; edge_ids(t, E, src, dst, ei, ej);
#pragma unroll
  for (int h = 0; h < 2; ++h) {
    float a = ai[ei*2 + h] + aj[ej*2 + h];
    a = a > 0.0f ? a : NEG_ATT * a;
    atomicMax(&amax[ei*2 + h], a);            // GLOBAL_ATOMIC_MAX_NUM_F32
  }
}

__global__ void k_fixup(float* __restrict__ amax, int N) {
  int i = blockIdx.x * blockDim.x + threadIdx.x;
  if (i >= N * 2) return;
  float v = amax[i];
  amax[i] = __builtin_isfinite(v) ? v : 0.0f;
}

// ---------------- ex = exp(alpha - amax[ei]); den = segment_sum(ex) ----------
__global__ void k_exp(const long long* __restrict__ src, const long long* __restrict__ dst,
                      const float* __restrict__ ai, const float* __restrict__ aj,
                      const float* __restrict__ amax, float* __restrict__ ex,
                      float* __restrict__ den, int E, int N) {
  int t = blockIdx.x * blockDim.x + threadIdx.x;
  if (t >= E + N) return;
  int ei, ej; edge_ids(t, E, src, dst, ei, ej);
#pragma unroll
  for (int h = 0; h < 2; ++h) {
    float a = ai[ei*2 + h] + aj[ej*2 + h];
    a = a > 0.0f ? a : NEG_ATT * a;
    float e = expf(a - amax[ei*2 + h]);
    ex[(size_t)t*2 + h] = e;
    atomicAdd(&den[ei*2 + h], e);
  }
}

// ---------------- supp[ei] += xt[ej] * alpha  (one wave per edge) ------------
__global__ __launch_bounds__(256) void k_scatter(const long long* __restrict__ src,
    const long long* __restrict__ dst, const float* __restrict__ xt,
    const float* __restrict__ ex, const float* __restrict__ den,
    float* __restrict__ supp, int E, int N) {
  int wid = (blockIdx.x * blockDim.x + threadIdx.x) >> 5;
  if (wid >= E + N) return;
  int lane = threadIdx.x & 31;
  int ei, ej; edge_ids(wid, E, src, dst, ei, ej);
  int head = lane >> 4;                       // channels 0-63 head0, 64-127 head1
  float alpha = ex[(size_t)wid*2 + head] / (den[ei*2 + head] + 1e-16f);
  float4 xv = ((const float4*)(xt + (size_t)ej * 128))[lane];
  float* d = supp + (size_t)ei * 128 + lane * 4;
  atomicAdd(d + 0, xv.x * alpha);
  atomicAdd(d + 1, xv.y * alpha);
  atomicAdd(d + 2, xv.z * alpha);
  atomicAdd(d + 3, xv.w * alpha);
}

// ------- mean heads -> proj(expmap0) -> leaky(logmap0) -> proj(expmap0) ------
__global__ __launch_bounds__(256) void k_final(const float* __restrict__ supp,
                                               float* __restrict__ out, int N) {
  int node = blockIdx.x * 8 + (threadIdx.x >> 5);
  if (node >= N) return;
  int lane = threadIdx.x & 31;
  int d = lane * 2;
  const float* sp = supp + (size_t)node * 128;
  float s0 = 0.5f * (sp[d]     + sp[64 + d]);
  float s1 = 0.5f * (sp[d + 1] + sp[64 + d + 1]);
  float n2 = wred(s0*s0 + s1*s1);
  float n  = fmaxf(sqrtf(n2), MIN_NORM_C);
  float e1 = tanhf(n) / n;                               // expmap0
  float hn = fmaxf(tanhf(n), MIN_NORM_C);
  float p1 = hn > MAXNORM_C ? MAXNORM_C / hn : 1.0f;     // proj
  float h0 = s0*e1*p1, h1 = s1*e1*p1;
  float nn = fmaxf(hn * p1, MIN_NORM_C);
  float ls = artanh_c(nn) / nn;                          // logmap0
  float t0 = h0 * ls, t1 = h1 * ls;
  t0 = t0 > 0.0f ? t0 : NEG_ACT * t0;                    // leaky_relu
  t1 = t1 > 0.0f ? t1 : NEG_ACT * t1;
  float tn2 = wred(t0*t0 + t1*t1);
  float tn  = fmaxf(sqrtf(tn2), MIN_NORM_C);
  float e2  = tanhf(tn) / tn;                            // expmap0
  float on  = fmaxf(tanhf(tn), MIN_NORM_C);
  float p2  = on > MAXNORM_C ? MAXNORM_C / on : 1.0f;    // proj
  ((float2*)(out + (size_t)node * 64))[lane] = make_float2(t0*e2*p2, t1*e2*p2);
}

extern "C" void kernel_launch(void* const* d_in, const int* in_sizes, int n_in,
                              void* d_out, int out_size, void* d_ws, size_t ws_size,
                              hipStream_t stream) {
  const float*      x      = (const float*)d_in[0];
  const long long*  edge   = (const long long*)d_in[1];  // int64 [2, E]
  const float*      weight = (const float*)d_in[2];
  const float*      bias   = (const float*)d_in[3];
  const float*      att_i  = (const float*)d_in[4];
  const float*      att_j  = (const float*)d_in[5];
  int N = in_sizes[0] / 128;
  int E = in_sizes[1] / 2;
  int T = E + N;
  const long long* src = edge;       // targets (ei)
  const long long* dst = edge + E;   // sources (ej)

  float* W = (float*)d_ws;
  size_t off = 0;
  float* mx_supp = W + off; off += (size_t)N * 128;   // mx, later reused as supp
  float* xt      = W + off; off += (size_t)N * 128;
  float* ai      = W + off; off += (size_t)N * 2;
  float* aj      = W + off; off += (size_t)N * 2;
  float* amax    = W + off; off += (size_t)N * 2;
  float* den     = W + off; off += (size_t)N * 2;
  float* ex      = W + off; off += (size_t)T * 2;
  float* hb      = W + off; off += 129;

  float* out = (float*)d_out;

  k_bias    <<<1, 128, 0, stream>>>(bias, hb);
  k_matmul  <<<(N + 15) / 16, 256, 0, stream>>>(x, weight, mx_supp, N);
  k_node    <<<(N + 7) / 8, 256, 0, stream>>>(x, mx_supp, hb, att_i, att_j,
                                              xt, ai, aj, N);
  k_init    <<<(N * 128 + 255) / 256, 256, 0, stream>>>(mx_supp, den, amax, N);
  k_edge_max<<<(T + 255) / 256, 256, 0, stream>>>(src, dst, ai, aj, amax, E, N);
  k_fixup   <<<(N * 2 + 255) / 256, 256, 0, stream>>>(amax, N);
  k_exp     <<<(T + 255) / 256, 256, 0, stream>>>(src, dst, ai, aj, amax, ex, den, E, N);
  k_scatter <<<(T + 7) / 8, 256, 0, stream>>>(src, dst, xt, ex, den, mx_supp, E, N);
  k_final   <<<(N + 7) / 8, 256, 0, stream>>>(mx_supp, out, N);
}